// QGCN_55817394979003
// MI455X (gfx1250) — compile-verified
//
#include <hip/hip_runtime.h>
#include <hip/hip_bf16.h>

typedef __attribute__((ext_vector_type(2))) float v2f;
typedef __attribute__((ext_vector_type(8))) float v8f;

#define DF 16
#define HDIM 64
#define CDIM 10

static __device__ __forceinline__ int wave_id_in_grid() {
    return (int)(blockIdx.x * (blockDim.x >> 5)) + (int)(threadIdx.x >> 5);
}

// ---------------- degree / dinv ----------------
__global__ void qgcn_deg_init(float* deg, int n) {
    int i = blockIdx.x * blockDim.x + threadIdx.x;
    if (i < n) deg[i] = 1.0f;   // self-loop weight
}

__global__ void qgcn_deg_acc(const int* __restrict__ col, const float* __restrict__ ea,
                             float* deg, int e) {
    int i = blockIdx.x * blockDim.x + threadIdx.x;
    if (i < e) {
        unsafeAtomicAdd(&deg[col[i]], ea[i]);
    }
}

__global__ void qgcn_dinv(float* d, int n) {
    int i = blockIdx.x * blockDim.x + threadIdx.x;
    if (i < n) {
        float v = d[i];
        d[i] = (v > 0.0f) ? rsqrtf(v) : 0.0f;
    }
}

// ---------------- lin: out = h @ W^T (D=16), WMMA f32 16x16x4 ----------------
// One wave per 16-row tile. A: lane<16 holds row m0+ln with K={k0,k0+1};
// lanes 16-31 hold same rows with K={k0+2,k0+3}. B[k][n] = W[n*16+k] with the
// mirrored striping. D: VGPR j -> row m0+j (lanes 0-15) / m0+j+8 (lanes 16-31).
__global__ __launch_bounds__(128) void qgcn_lin16(const float* __restrict__ h,
                                                  const float* __restrict__ W,
                                                  float* __restrict__ out, int ntiles) {
    int wave = wave_id_in_grid();
    if (wave >= ntiles) return;
    int lane = threadIdx.x & 31;
    int half = lane >> 4;
    int ln   = lane & 15;
    int m0   = wave * 16;

    const float* hrow = h + (size_t)(m0 + ln) * DF;
    const float* wrow = W + (size_t)ln * DF;

    v8f c = {};
#pragma unroll
    for (int kk = 0; kk < 4; ++kk) {
        int kb = kk * 4 + 2 * half;
        v2f a, b;
        a[0] = hrow[kb];     a[1] = hrow[kb + 1];
        b[0] = wrow[kb];     b[1] = wrow[kb + 1];
        c = __builtin_amdgcn_wmma_f32_16x16x4_f32(false, a, false, b, (short)0, c,
                                                  false, false);
    }
    float* orow = out + (size_t)m0 * DF;
#pragma unroll
    for (int j = 0; j < 8; ++j)
        orow[(size_t)(j + 8 * half) * DF + ln] = c[j];
}

// ---------------- agg init with self-loop contribution ----------------
__global__ void qgcn_agg_init(const float* __restrict__ lo, const float* __restrict__ dinv,
                              float* __restrict__ agg, int n) {
    int t = blockIdx.x * blockDim.x + threadIdx.x;
    if (t < n * DF) {
        int i = t >> 4;
        float di = dinv[i];
        agg[t] = di * di * lo[t];
    }
}

// ---------------- edge scatter: 4 threads per edge, float4 payload ----------------
__global__ void qgcn_scatter(const int* __restrict__ row, const int* __restrict__ col,
                             const float* __restrict__ ea, const float* __restrict__ dinv,
                             const float* __restrict__ lo, float* __restrict__ agg, int e) {
    int tid = blockIdx.x * blockDim.x + threadIdx.x;
    int idx = tid >> 2;
    int q   = tid & 3;
    if (idx < e) {
        if (idx + 32768 < e) {
            __builtin_prefetch(ea + idx + 32768, 0, 1);      // global_prefetch_b8
            __builtin_prefetch(row + idx + 32768, 0, 1);
        }
        int r = row[idx];
        int c = col[idx];
        float nrm = dinv[r] * ea[idx] * dinv[c];
        float4 v = reinterpret_cast<const float4*>(lo)[(size_t)r * 4 + q];
        float* dst = agg + (size_t)c * DF + q * 4;
        unsafeAtomicAdd(dst + 0, nrm * v.x);
        unsafeAtomicAdd(dst + 1, nrm * v.y);
        unsafeAtomicAdd(dst + 2, nrm * v.z);
        unsafeAtomicAdd(dst + 3, nrm * v.w);
    }
}

// ------- node update: WMMA matmul + tanh + conv_b + LeakyReLU + LayerNorm -------
__global__ __launch_bounds__(128) void qgcn_node_update(const float* __restrict__ agg,
                                                        const float* __restrict__ Wn,
                                                        const float* __restrict__ nb,
                                                        const float* __restrict__ cb,
                                                        const float* __restrict__ lg,
                                                        const float* __restrict__ lb,
                                                        float* __restrict__ hout, int ntiles) {
    int wave = wave_id_in_grid();
    if (wave >= ntiles) return;
    int lane = threadIdx.x & 31;
    int half = lane >> 4;
    int ln   = lane & 15;
    int m0   = wave * 16;

    const float* arow = agg + (size_t)(m0 + ln) * DF;
    const float* wrow = Wn + (size_t)ln * DF;

    v8f c = {};
#pragma unroll
    for (int kk = 0; kk < 4; ++kk) {
        int kb = kk * 4 + 2 * half;
        v2f a, b;
        a[0] = arow[kb]; a[1] = arow[kb + 1];
        b[0] = wrow[kb]; b[1] = wrow[kb + 1];
        c = __builtin_amdgcn_wmma_f32_16x16x4_f32(false, a, false, b, (short)0, c,
                                                  false, false);
    }

    float nbv = nb[ln], cbv = cb[ln], gv = lg[ln], bv = lb[ln];
#pragma unroll
    for (int j = 0; j < 8; ++j) {
        float v = tanhf(c[j] + nbv) + cbv;
        v = (v > 0.0f) ? v : 0.2f * v;                 // LeakyReLU(0.2)
        // LayerNorm across the 16 lanes of this half (one matrix row)
        float s = v;
        s += __shfl_xor(s, 1); s += __shfl_xor(s, 2);
        s += __shfl_xor(s, 4); s += __shfl_xor(s, 8);
        float mu = s * (1.0f / 16.0f);
        float d  = v - mu;
        float q  = d * d;
        q += __shfl_xor(q, 1); q += __shfl_xor(q, 2);
        q += __shfl_xor(q, 4); q += __shfl_xor(q, 8);
        float var = q * (1.0f / 16.0f);
        float r = rsqrtf(var + 1e-5f);
        hout[(size_t)(m0 + j + 8 * half) * DF + ln] = d * r * gv + bv;
    }
}

// ------- classifier: relu(h @ w1^T + b1) @ w2^T + b2, fused, LDS staging -------
__global__ __launch_bounds__(128) void qgcn_classifier(const float* __restrict__ h,
                                                       const float* __restrict__ w1,
                                                       const float* __restrict__ b1,
                                                       const float* __restrict__ w2,
                                                       const float* __restrict__ b2,
                                                       float* __restrict__ out, int ntiles) {
    __shared__ float h1s[4][16][HDIM + 1];   // +1 pad: kill LDS bank conflicts
    int wslot = threadIdx.x >> 5;
    int wave  = wave_id_in_grid();
    if (wave >= ntiles) return;
    int lane = threadIdx.x & 31;
    int half = lane >> 4;
    int ln   = lane & 15;
    int m0   = wave * 16;

    const float* hrow = h + (size_t)(m0 + ln) * DF;

    // h1 = relu(h @ w1^T + b1): 4 column tiles of 16, K=16 in 4 WMMA steps each
    v8f acc[4] = {{}, {}, {}, {}};
#pragma unroll
    for (int kk = 0; kk < 4; ++kk) {
        int kb = kk * 4 + 2 * half;
        v2f a;
        a[0] = hrow[kb]; a[1] = hrow[kb + 1];
#pragma unroll
        for (int t = 0; t < 4; ++t) {
            const float* wr = w1 + (size_t)(16 * t + ln) * DF;
            v2f b;
            b[0] = wr[kb]; b[1] = wr[kb + 1];
            acc[t] = __builtin_amdgcn_wmma_f32_16x16x4_f32(false, a, false, b, (short)0,
                                                           acc[t], false, false);
        }
    }
#pragma unroll
    for (int t = 0; t < 4; ++t) {
        float b1v = b1[16 * t + ln];
#pragma unroll
        for (int j = 0; j < 8; ++j) {
            float v = acc[t][j] + b1v;
            h1s[wslot][j + 8 * half][16 * t + ln] = (v > 0.0f) ? v : 0.0f;
        }
    }
    // DS ops are in-order within a wave: store->load ordering is safe.

    // out = h1 @ w2^T + b2 : K=64 in 16 WMMA steps; N=10 padded to 16 with zeros
    v8f c2 = {};
#pragma unroll
    for (int kk = 0; kk < 16; ++kk) {
        int kb = kk * 4 + 2 * half;
        v2f a, b;
        a[0] = h1s[wslot][ln][kb];
        a[1] = h1s[wslot][ln][kb + 1];
        float w0 = 0.0f, w1v = 0.0f;
        if (ln < CDIM) {
            const float* wr = w2 + (size_t)ln * HDIM;
            w0 = wr[kb]; w1v = wr[kb + 1];
        }
        b[0] = w0; b[1] = w1v;
        c2 = __builtin_amdgcn_wmma_f32_16x16x4_f32(false, a, false, b, (short)0, c2,
                                                   false, false);
    }
    if (ln < CDIM) {
        float bb = b2[ln];
#pragma unroll
        for (int j = 0; j < 8; ++j)
            out[(size_t)(m0 + j + 8 * half) * CDIM + ln] = c2[j] + bb;
    }
}

extern "C" void kernel_launch(void* const* d_in, const int* in_sizes, int n_in,
                              void* d_out, int out_size, void* d_ws, size_t ws_size,
                              hipStream_t stream) {
    const float* x      = (const float*)d_in[0];
    const int*   ei     = (const int*)d_in[1];
    const float* ea     = (const float*)d_in[2];
    // d_in[3] = batch (unused: single graph, mean-pool absent in reference)
    const float* lin_w  = (const float*)d_in[4];
    const float* conv_b = (const float*)d_in[5];
    const float* node_w = (const float*)d_in[6];
    const float* node_b = (const float*)d_in[7];
    const float* ln_g   = (const float*)d_in[8];
    const float* ln_b   = (const float*)d_in[9];
    const float* w1     = (const float*)d_in[10];
    const float* b1     = (const float*)d_in[11];
    const float* w2     = (const float*)d_in[12];
    const float* b2     = (const float*)d_in[13];
    float* out = (float*)d_out;

    const int n = in_sizes[0] / DF;   // 100000
    const int e = in_sizes[2];        // 3200000
    const int L = in_sizes[4] / (DF * DF);   // 2

    const int* row = ei;
    const int* col = ei + e;

    float* ws   = (float*)d_ws;
    float* dinv = ws;                           // n floats
    float* bufA = ws + n;                       // n*DF
    float* bufB = bufA + (size_t)n * DF;        // n*DF

    auto cdiv = [](int a, int b) { return (a + b - 1) / b; };
    const int ntiles = cdiv(n, 16);             // n is a multiple of 16
    const int wblocks = cdiv(ntiles, 4);        // 4 waves / 128-thread block

    // normalization coefficients
    qgcn_deg_init<<<cdiv(n, 256), 256, 0, stream>>>(dinv, n);
    qgcn_deg_acc<<<cdiv(e, 256), 256, 0, stream>>>(col, ea, dinv, e);
    qgcn_dinv<<<cdiv(n, 256), 256, 0, stream>>>(dinv, n);

    const float* h = x;
    float* ping = bufA;
    float* pong = bufB;
    for (int l = 0; l < L; ++l) {
        qgcn_lin16<<<wblocks, 128, 0, stream>>>(h, lin_w + (size_t)l * DF * DF, ping, ntiles);
        qgcn_agg_init<<<cdiv(n * DF, 256), 256, 0, stream>>>(ping, dinv, pong, n);
        qgcn_scatter<<<cdiv(e * 4, 256), 256, 0, stream>>>(row, col, ea, dinv, ping, pong, e);
        qgcn_node_update<<<wblocks, 128, 0, stream>>>(pong, node_w + (size_t)l * DF * DF,
                                                      node_b + l * DF, conv_b + l * DF,
                                                      ln_g + l * DF, ln_b + l * DF,
                                                      ping, ntiles);
        h = ping;
        float* t = ping; ping = pong; pong = t;
    }

    qgcn_classifier<<<wblocks, 128, 0, stream>>>(h, w1, b1, w2, b2, out, ntiles);
}